// AttentionWithRoPE_44555990729361
// MI455X (gfx1250) — compile-verified
//
#include <hip/hip_runtime.h>
#include <hip/hip_bf16.h>

typedef __attribute__((ext_vector_type(2))) float v2f;
typedef __attribute__((ext_vector_type(4))) float v4f;
typedef __attribute__((ext_vector_type(8))) float v8f;

#define BATCH 16
#define HID   4096
#define NH    32
#define HD    128
#define BS    16
#define MAXB  64
#define SMAX  (MAXB * BS)      /* 1024 */
#define QKV_N (3 * HID)        /* 12288 */
#define SCALE 0.08838834764831845f  /* 1/sqrt(128) */

// ---------------------------------------------------------------------------
// GEMM: C[16, N] = A[16, 4096] @ W[N, 4096]^T  via V_WMMA_F32_16X16X4_F32
// One wave (32 lanes) per 16-wide tile of N; K accumulated in steps of 4.
//
// A 16x4 f32 layout : lane l holds A[l&15][kpair], A[l&15][kpair+1],
//                     kpair = (l>>4)*2  -> contiguous float2 in row-major A.
// B 4x16  f32 layout: lane l holds B[kpair][l&15], B[kpair+1][l&15]
//                     = W[n][k], W[n][k+1]        -> contiguous float2 in W.
// D 16x16 f32 layout: VGPR j, lanes 0-15 -> M=j, lanes 16-31 -> M=j+8.
// ---------------------------------------------------------------------------
__global__ __launch_bounds__(32)
void gemm16_wmma_f32(const float* __restrict__ A, const float* __restrict__ W,
                     float* __restrict__ C, int N) {
    const int nbase = blockIdx.x * 16;
    const int lane  = threadIdx.x;
    const int idx16 = lane & 15;          // M for A, N-within-tile for B
    const int kpair = (lane >> 4) * 2;    // 0 or 2

    const float* aptr = A + idx16 * HID + kpair;
    const float* bptr = W + (size_t)(nbase + idx16) * HID + kpair;

    v8f acc = {};
    for (int k = 0; k < HID; k += 16) {
        __builtin_prefetch(bptr + k + 256, 0, 1);
#pragma unroll
        for (int u = 0; u < 4; ++u) {
            v2f a = *(const v2f*)(aptr + k + u * 4);
            v2f b = *(const v2f*)(bptr + k + u * 4);
            acc = __builtin_amdgcn_wmma_f32_16x16x4_f32(
                false, a, false, b, (short)0, acc, false, false);
        }
    }

    const int m0 = (lane >> 4) * 8;
    const int n  = nbase + idx16;
#pragma unroll
    for (int j = 0; j < 8; ++j) {
        C[(size_t)(m0 + j) * N + n] = acc[j];
    }
}

// ---------------------------------------------------------------------------
// RoPE (NeoX style, full rotation over HD): one thread per (token, head, pair)
// ---------------------------------------------------------------------------
__global__ void rope_kernel(const float* __restrict__ qkv,
                            const float* __restrict__ cs_cache,
                            const int*   __restrict__ pos_ids,
                            float* __restrict__ q_out,
                            float* __restrict__ k_out) {
    int idx = blockIdx.x * blockDim.x + threadIdx.x;   // B*NH*64 total
    if (idx >= BATCH * NH * 64) return;
    int t   = idx / (NH * 64);
    int rem = idx % (NH * 64);
    int h   = rem / 64;
    int i   = rem % 64;

    int   pos = pos_ids[t];
    float c = cs_cache[(size_t)pos * HD + i];
    float s = cs_cache[(size_t)pos * HD + 64 + i];

    const float* qrow = qkv + (size_t)t * QKV_N;           // q part
    const float* krow = qrow + HID;                        // k part

    float q1 = qrow[h * HD + i], q2 = qrow[h * HD + 64 + i];
    q_out[(size_t)t * HID + h * HD + i]      = q1 * c - q2 * s;
    q_out[(size_t)t * HID + h * HD + 64 + i] = q2 * c + q1 * s;

    float k1 = krow[h * HD + i], k2 = krow[h * HD + 64 + i];
    k_out[(size_t)t * HID + h * HD + i]      = k1 * c - k2 * s;
    k_out[(size_t)t * HID + h * HD + 64 + i] = k2 * c + k1 * s;
}

// ---------------------------------------------------------------------------
// Paged attention decode: one block (128 thr = 4 waves) per (batch, head).
// New k/v are substituted at s == ctx-1 (== slots[b] in the gathered view).
// ---------------------------------------------------------------------------
__global__ __launch_bounds__(128)
void attn_kernel(const float* __restrict__ q_rope,
                 const float* __restrict__ k_rope,
                 const float* __restrict__ qkv,          // v part lives here
                 const float* __restrict__ key_cache,
                 const float* __restrict__ value_cache,
                 const int*   __restrict__ block_tables,
                 const int*   __restrict__ ctx_lens,
                 float* __restrict__ attn_out) {
    __shared__ float sc[SMAX];
    __shared__ float qsh[HD];
    __shared__ float red[8];

    const int b   = blockIdx.x / NH;
    const int h   = blockIdx.x % NH;
    const int tid = threadIdx.x;
    const int ctx  = ctx_lens[b];
    const int last = ctx - 1;

    if (tid < HD) qsh[tid] = q_rope[(size_t)b * HID + h * HD + tid];
    __syncthreads();

    // ---- pass 1: scores ----
    for (int s = tid; s < ctx; s += 128) {
        const float* kptr;
        if (s == last) {
            kptr = k_rope + (size_t)b * HID + h * HD;
        } else {
            int blk = block_tables[b * MAXB + (s >> 4)];
            kptr = key_cache + ((size_t)(blk * BS + (s & 15)) * NH + h) * HD;
        }
        float acc = 0.f;
#pragma unroll
        for (int d = 0; d < HD; d += 4) {
            v4f kv = *(const v4f*)(kptr + d);
            acc += kv.x * qsh[d] + kv.y * qsh[d + 1] +
                   kv.z * qsh[d + 2] + kv.w * qsh[d + 3];
        }
        sc[s] = acc * SCALE;
    }
    __syncthreads();

    // ---- max reduce (wave32 shuffles + LDS) ----
    float m = -1e30f;
    for (int s = tid; s < ctx; s += 128) m = fmaxf(m, sc[s]);
#pragma unroll
    for (int off = 16; off > 0; off >>= 1) m = fmaxf(m, __shfl_xor(m, off, 32));
    if ((tid & 31) == 0) red[tid >> 5] = m;
    __syncthreads();
    if (tid == 0)
        red[4] = fmaxf(fmaxf(red[0], red[1]), fmaxf(red[2], red[3]));
    __syncthreads();
    m = red[4];
    __syncthreads();

    // ---- exp + sum reduce ----
    float lsum = 0.f;
    for (int s = tid; s < ctx; s += 128) {
        float e = __expf(sc[s] - m);
        sc[s] = e;
        lsum += e;
    }
#pragma unroll
    for (int off = 16; off > 0; off >>= 1) lsum += __shfl_xor(lsum, off, 32);
    if ((tid & 31) == 0) red[tid >> 5] = lsum;
    __syncthreads();
    if (tid == 0) red[4] = red[0] + red[1] + red[2] + red[3];
    __syncthreads();
    const float total = red[4];

    // ---- pass 2: weighted V (thread owns one d, coalesced 512B per step) ----
    const int d = tid;   // 128 threads == HD
    float acc = 0.f;
    for (int s = 0; s < ctx; ++s) {
        const float* vptr;
        if (s == last) {
            vptr = qkv + (size_t)b * QKV_N + 2 * HID + h * HD;
        } else {
            int blk = block_tables[b * MAXB + (s >> 4)];
            vptr = value_cache + ((size_t)(blk * BS + (s & 15)) * NH + h) * HD;
        }
        acc += sc[s] * vptr[d];
    }
    attn_out[(size_t)b * HID + h * HD + d] = acc / total;
}

// ---------------------------------------------------------------------------
extern "C" void kernel_launch(void* const* d_in, const int* in_sizes, int n_in,
                              void* d_out, int out_size, void* d_ws, size_t ws_size,
                              hipStream_t stream) {
    const float* hidden      = (const float*)d_in[0];   // [16,1,4096]
    const float* qkv_w       = (const float*)d_in[1];   // [12288,4096]
    const float* out_w       = (const float*)d_in[2];   // [4096,4096]
    const float* cs_cache    = (const float*)d_in[3];   // [2048,128]
    const float* key_cache   = (const float*)d_in[4];   // [1024,16,32,128]
    const float* value_cache = (const float*)d_in[5];   // [1024,16,32,128]
    const int*   pos_ids     = (const int*)d_in[6];     // [16,1]
    const int*   block_tabs  = (const int*)d_in[7];     // [16,64]
    // d_in[8] = slots (unused: implied by ctx-1 in gathered view)
    const int*   ctx_lens    = (const int*)d_in[9];     // [16]
    float*       out         = (float*)d_out;           // [16,1,4096]

    float* ws     = (float*)d_ws;
    float* qkv    = ws;                               // 16*12288
    float* q_rope = qkv    + (size_t)BATCH * QKV_N;   // 16*4096
    float* k_rope = q_rope + (size_t)BATCH * HID;     // 16*4096
    float* attn   = k_rope + (size_t)BATCH * HID;     // 16*4096

    // 1) QKV projection (WMMA f32): [16,12288]
    gemm16_wmma_f32<<<QKV_N / 16, 32, 0, stream>>>(hidden, qkv_w, qkv, QKV_N);

    // 2) RoPE on q and k
    rope_kernel<<<(BATCH * NH * 64 + 255) / 256, 256, 0, stream>>>(
        qkv, cs_cache, pos_ids, q_rope, k_rope);

    // 3) Paged attention, one block per (b,h)
    attn_kernel<<<BATCH * NH, 128, 0, stream>>>(
        q_rope, k_rope, qkv, key_cache, value_cache, block_tabs, ctx_lens, attn);

    // 4) Output projection (WMMA f32): [16,4096]
    gemm16_wmma_f32<<<HID / 16, 32, 0, stream>>>(attn, out_w, out, HID);
}